// Attention_55740085567993
// MI455X (gfx1250) — compile-verified
//
#include <hip/hip_runtime.h>
#include <math.h>

// ---------------------------------------------------------------------------
// Paged-attention decode (RMSNorm + NeoX RoPE + GQA flash attention), fp32.
// MI455X roofline: K+V stream = 536 MB -> ~23us @ 23.3 TB/s; useful math is
// ~1.07 GFLOP -> ~46 TFLOPS needed. Bandwidth-bound, so we keep fp32 end to
// end and use V_WMMA_F32_16X16X4_F32 (exact fp32 matrix op) for both GEMMs.
// Half-wave (xor-16) shuffles use v_permlanex16_b32 (VALU) instead of
// ds_bpermute to keep the softmax chain off the LDS pipe.
// ---------------------------------------------------------------------------

#define NUM_SEQS        64
#define NUM_HEADS       32
#define NUM_KV_HEADS    8
#define GQA             4          // NUM_HEADS / NUM_KV_HEADS
#define HEAD_DIM        128
#define BLOCK_SZ        16
#define MAX_SEQLEN_K    1024
#define BLOCKS_PER_SEQ  64
#define ATTN_SCALE      0.08838834764831845f   // 1/sqrt(128)
#define NEG_BIG         (-3.0e38f)

typedef __attribute__((ext_vector_type(2))) float v2f;
typedef __attribute__((ext_vector_type(8))) float v8f;

// xor-16 half-wave swap as a VALU op (v_permlanex16_b32, identity selects)
__device__ __forceinline__ float xor16f(float v)
{
    int i = __builtin_bit_cast(int, v);
    int r = __builtin_amdgcn_permlanex16(i, i, 0x76543210, 0xfedcba98, false, false);
    return __builtin_bit_cast(float, r);
}

// ---------------------------------------------------------------------------
// Kernel 1: RMSNorm + RoPE for q (2048 rows) and k (512 rows).
// One wave per 128-dim row; lane l owns dims {2l,2l+1, 64+2l, 64+2l+1}
// so the NeoX half-rotation is lane-local.
// ---------------------------------------------------------------------------
__global__ void qk_prep_kernel(const float* __restrict__ q,
                               const float* __restrict__ k,
                               const float* __restrict__ qw,
                               const float* __restrict__ kw,
                               const float* __restrict__ cos_cache,
                               const float* __restrict__ sin_cache,
                               const int*   __restrict__ position,
                               float* __restrict__ qn,
                               float* __restrict__ kn)
{
    const int wave = threadIdx.x >> 5;
    const int lane = threadIdx.x & 31;
    const int row  = blockIdx.x * 4 + wave;      // 0 .. 2559

    const float* src; const float* w; float* dst; int s;
    if (row < NUM_SEQS * NUM_HEADS) {            // q row
        s   = row >> 5;
        src = q  + (size_t)row * HEAD_DIM;
        w   = qw;
        dst = qn + (size_t)row * HEAD_DIM;
    } else {                                     // k row
        const int r2 = row - NUM_SEQS * NUM_HEADS;
        s   = r2 >> 3;
        src = k  + (size_t)r2 * HEAD_DIM;
        w   = kw;
        dst = kn + (size_t)r2 * HEAD_DIM;
    }

    const int i = lane * 2;                      // 0..62 within each half
    float x1a = src[i],      x1b = src[i + 1];
    float x2a = src[i + 64], x2b = src[i + 65];

    float ss = x1a * x1a + x1b * x1b + x2a * x2a + x2b * x2b;
    #pragma unroll
    for (int off = 16; off >= 1; off >>= 1) ss += __shfl_xor(ss, off);
    const float rinv = rsqrtf(ss * (1.0f / HEAD_DIM) + 1e-6f);

    x1a *= rinv * w[i];      x1b *= rinv * w[i + 1];
    x2a *= rinv * w[i + 64]; x2b *= rinv * w[i + 65];

    const int pos = position[s];
    const float* cc = cos_cache + (size_t)pos * (HEAD_DIM / 2);
    const float* sc = sin_cache + (size_t)pos * (HEAD_DIM / 2);
    const float ca = cc[i], cb = cc[i + 1];
    const float sa = sc[i], sb = sc[i + 1];

    dst[i]      = x1a * ca - x2a * sa;
    dst[i + 1]  = x1b * cb - x2b * sb;
    dst[i + 64] = x2a * ca + x1a * sa;
    dst[i + 65] = x2b * cb + x1b * sb;
}

// ---------------------------------------------------------------------------
// Kernel 2: flash attention per (kv head, seq). 4 waves split key tiles.
//   GEMM1: S^T = K_tile(16x128) x Q^T(128x16)  via 32 chained f32 WMMAs
//   GEMM2: out^T += V^T(16x16-chunks) x P^T    via 8x4 f32 WMMAs / tile
// The updated token (index ctx-1) is sourced from kn / raw v, so the input
// caches are never mutated.
// ---------------------------------------------------------------------------
__launch_bounds__(128)
__global__ void paged_attn_kernel(const float* __restrict__ qn,
                                  const float* __restrict__ kn,
                                  const float* __restrict__ vnew,
                                  const float* __restrict__ k_cache,
                                  const float* __restrict__ v_cache,
                                  const int*   __restrict__ block_tables,
                                  const int*   __restrict__ context_lens,
                                  float* __restrict__ out)
{
    __shared__ float lds_acc[4 * GQA * HEAD_DIM];   // 8 KB partial outputs
    __shared__ float lds_m[4 * GQA];
    __shared__ float lds_l[4 * GQA];

    const int h    = blockIdx.x;          // kv head
    const int s    = blockIdx.y;          // sequence
    const int wave = threadIdx.x >> 5;
    const int lane = threadIdx.x & 31;
    const int half = lane >> 4;           // 0 or 1
    const int lidx = lane & 15;
    const int ctx  = context_lens[s];

    // ---- Q as B-operand (heads in columns, SCALE folded in) --------------
    v2f b_q[32];
    {
        const float* qbase = qn + ((size_t)s * NUM_HEADS + h * GQA) * HEAD_DIM;
        #pragma unroll
        for (int c = 0; c < 32; ++c) {
            float bx = 0.0f, by = 0.0f;
            if (lidx < GQA) {
                const float* qr = qbase + (size_t)lidx * HEAD_DIM + 4 * c + 2 * half;
                bx = qr[0] * ATTN_SCALE;
                by = qr[1] * ATTN_SCALE;
            }
            b_q[c].x = bx; b_q[c].y = by;
        }
    }

    float m_st = NEG_BIG;
    float l_st = 0.0f;
    v8f acc[8];
    #pragma unroll
    for (int dt = 0; dt < 8; ++dt) { v8f z = {}; acc[dt] = z; }

    const int    ntiles = (ctx + 15) >> 4;
    const int*   bt     = block_tables + s * BLOCKS_PER_SEQ;
    const float* knrow  = kn   + ((size_t)s * NUM_KV_HEADS + h) * HEAD_DIM;
    const float* vnrow  = vnew + ((size_t)s * NUM_KV_HEADS + h) * HEAD_DIM;

    for (int t = wave; t < ntiles; t += 4) {
        const int base = t * 16;
        const int blk  = bt[t];

        // prefetch next tile's K rows (global_prefetch_b8)
        if (t + 4 < ntiles) {
            const float* pf = k_cache +
                (((size_t)bt[t + 4] * BLOCK_SZ + lidx) * NUM_KV_HEADS + h) * HEAD_DIM;
            __builtin_prefetch(pf, 0, 1);
        }

        // K row for this lane's key (substitute new token at ctx-1)
        const int nk = base + lidx;
        const float* krow = (nk == ctx - 1)
            ? knrow
            : k_cache + (((size_t)blk * BLOCK_SZ + lidx) * NUM_KV_HEADS + h) * HEAD_DIM;

        // ---- GEMM1: S^T = K x Q^T, accumulate over d in chunks of 4 ------
        v8f st = {};
        #pragma unroll
        for (int c = 0; c < 32; ++c) {
            v2f a;
            const float* kp = krow + 4 * c + 2 * half;
            a.x = kp[0]; a.y = kp[1];
            st = __builtin_amdgcn_wmma_f32_16x16x4_f32(
                     false, a, false, b_q[c], (short)0, st, false, false);
        }

        // ---- mask + online softmax (state per head = lane%16) ------------
        float p[8];
        float tmax = NEG_BIG;
        #pragma unroll
        for (int r = 0; r < 8; ++r) {
            const int n = base + r + 8 * half;
            const float sv = (n < ctx) ? st[r] : NEG_BIG;
            p[r] = sv;
            tmax = fmaxf(tmax, sv);
        }
        tmax = fmaxf(tmax, xor16f(tmax));
        const float m_new = fmaxf(m_st, tmax);
        const float alpha = __expf(m_st - m_new);
        float sum = 0.0f;
        #pragma unroll
        for (int r = 0; r < 8; ++r) {
            p[r] = __expf(p[r] - m_new);
            sum += p[r];
        }
        sum += xor16f(sum);
        l_st = l_st * alpha + sum;
        m_st = m_new;

        #pragma unroll
        for (int dt = 0; dt < 8; ++dt)
            #pragma unroll
            for (int r = 0; r < 8; ++r) acc[dt][r] *= alpha;

        // ---- build P^T B-operands (8 half-swap VALU permutes) ------------
        const float sp0 = xor16f(p[0]), sp1 = xor16f(p[1]);
        const float sp2 = xor16f(p[2]), sp3 = xor16f(p[3]);
        const float sp4 = xor16f(p[4]), sp5 = xor16f(p[5]);
        const float sp6 = xor16f(p[6]), sp7 = xor16f(p[7]);
        v2f b_p[4];
        b_p[0].x = half ? sp2 : p[0];  b_p[0].y = half ? sp3 : p[1];
        b_p[1].x = half ? sp6 : p[4];  b_p[1].y = half ? sp7 : p[5];
        b_p[2].x = half ? p[2] : sp0;  b_p[2].y = half ? p[3] : sp1;
        b_p[3].x = half ? p[6] : sp4;  b_p[3].y = half ? p[7] : sp5;

        // ---- V row pointers for this lane's keys -------------------------
        const float* vrowA[4];
        const float* vrowB[4];
        #pragma unroll
        for (int c = 0; c < 4; ++c) {
            const int key0 = 4 * c + 2 * half;
            const int n0 = base + key0;
            vrowA[c] = (n0 == ctx - 1) ? vnrow
                : v_cache + (((size_t)blk * BLOCK_SZ + key0) * NUM_KV_HEADS + h) * HEAD_DIM;
            vrowB[c] = (n0 + 1 == ctx - 1) ? vnrow
                : v_cache + (((size_t)blk * BLOCK_SZ + key0 + 1) * NUM_KV_HEADS + h) * HEAD_DIM;
        }

        // ---- GEMM2: out^T += V^T x P^T (coalesced V loads) ---------------
        #pragma unroll
        for (int dt = 0; dt < 8; ++dt) {
            const int d = dt * 16 + lidx;
            #pragma unroll
            for (int c = 0; c < 4; ++c) {
                v2f a;
                a.x = vrowA[c][d];
                a.y = vrowB[c][d];
                acc[dt] = __builtin_amdgcn_wmma_f32_16x16x4_f32(
                              false, a, false, b_p[c], (short)0, acc[dt], false, false);
            }
        }
    }

    // ---- cross-wave merge through LDS ------------------------------------
    if (lidx < GQA && half == 0) {
        lds_m[wave * GQA + lidx] = m_st;
        lds_l[wave * GQA + lidx] = l_st;
    }
    if (lidx < GQA) {
        float* dstl = lds_acc + ((size_t)wave * GQA + lidx) * HEAD_DIM;
        #pragma unroll
        for (int dt = 0; dt < 8; ++dt)
            #pragma unroll
            for (int r = 0; r < 8; ++r)
                dstl[dt * 16 + r + 8 * half] = acc[dt][r];
    }
    __syncthreads();

    float* outb = out + ((size_t)s * NUM_HEADS + h * GQA) * HEAD_DIM;
    for (int idx = threadIdx.x; idx < GQA * HEAD_DIM; idx += 128) {
        const int g = idx >> 7;
        const int d = idx & 127;
        const float m0 = lds_m[0 * GQA + g], m1 = lds_m[1 * GQA + g];
        const float m2 = lds_m[2 * GQA + g], m3 = lds_m[3 * GQA + g];
        const float mm = fmaxf(fmaxf(m0, m1), fmaxf(m2, m3));
        const float e0 = __expf(m0 - mm), e1 = __expf(m1 - mm);
        const float e2 = __expf(m2 - mm), e3 = __expf(m3 - mm);
        const float L  = e0 * lds_l[0 * GQA + g] + e1 * lds_l[1 * GQA + g]
                       + e2 * lds_l[2 * GQA + g] + e3 * lds_l[3 * GQA + g];
        const float vsum = e0 * lds_acc[(0 * GQA + g) * HEAD_DIM + d]
                         + e1 * lds_acc[(1 * GQA + g) * HEAD_DIM + d]
                         + e2 * lds_acc[(2 * GQA + g) * HEAD_DIM + d]
                         + e3 * lds_acc[(3 * GQA + g) * HEAD_DIM + d];
        outb[idx] = vsum / L;
    }
}

// ---------------------------------------------------------------------------
extern "C" void kernel_launch(void* const* d_in, const int* in_sizes, int n_in,
                              void* d_out, int out_size, void* d_ws, size_t ws_size,
                              hipStream_t stream)
{
    const float* q            = (const float*)d_in[0];
    const float* k            = (const float*)d_in[1];
    const float* v            = (const float*)d_in[2];
    const float* k_cache      = (const float*)d_in[3];
    const float* v_cache      = (const float*)d_in[4];
    const float* qw           = (const float*)d_in[5];
    const float* kw           = (const float*)d_in[6];
    const float* cos_cache    = (const float*)d_in[7];
    const float* sin_cache    = (const float*)d_in[8];
    const int*   position     = (const int*)d_in[9];
    /* d_in[10] slot_mapping: implied by position/block_tables, unused */
    const int*   block_tables = (const int*)d_in[11];
    const int*   context_lens = (const int*)d_in[12];
    float* out = (float*)d_out;

    float* qn = (float*)d_ws;                                   // 64*32*128 f32
    float* kn = qn + (size_t)NUM_SEQS * NUM_HEADS * HEAD_DIM;   // 64*8*128  f32

    const int rows   = NUM_SEQS * NUM_HEADS + NUM_SEQS * NUM_KV_HEADS; // 2560
    const int blocks = rows / 4;                                       // 640
    qk_prep_kernel<<<blocks, 128, 0, stream>>>(q, k, qw, kw, cos_cache, sin_cache,
                                               position, qn, kn);

    dim3 grid(NUM_KV_HEADS, NUM_SEQS);
    paged_attn_kernel<<<grid, 128, 0, stream>>>(qn, kn, v, k_cache, v_cache,
                                                block_tables, context_lens, out);
}